// HarmonicOscillator_57569741636337
// MI455X (gfx1250) — compile-verified
//
#include <hip/hip_runtime.h>
#include <hip/hip_bf16.h>
#include <stdint.h>

// ---------------- problem constants ----------------
#define NBATCH    32
#define L_FRAMES  800
#define FRAME     240
#define T_TOTAL   192000          // L_FRAMES * FRAME
#define NHARM     8
#define INV_SR    (1.0f / 24000.0f)
#define TWO_PI_F  6.28318530717958647692f

// ---------------- main kernel tiling ----------------
#define TPB   192                 // threads per block (6 wave32s)
#define SPT   4                   // samples per thread (float4)
#define TILE  (TPB * SPT)         // 768 samples; 192000 % 768 == 0

// ---- CDNA5 async global->LDS staging (gfx1250) ----
__device__ __forceinline__ void async_ld_b128(uint32_t lds_off, const void* gaddr) {
  // per-lane 16B: copied from the global address gaddr to the LDS address lds_off; tracked by ASYNCcnt
  asm volatile("global_load_async_to_lds_b128 %0, %1, off"
               :: "v"(lds_off), "v"(gaddr)
               : "memory");
}
__device__ __forceinline__ void wait_async0() {
#if __has_builtin(__builtin_amdgcn_s_wait_asynccnt)
  __builtin_amdgcn_s_wait_asynccnt(0);
#else
  asm volatile("s_wait_asynccnt 0x0" ::: "memory");
#endif
}

// =====================================================================
// Kernel 1: per-batch frame-boundary phase prefix (mod 1).
// Frame sum of f0_up/SR has closed form (30*A + 180*B + 30*C)/SR with
// A=f0[l-1],B=f0[l],C=f0[l+1] (edge-clamped). Hillis-Steele scan with
// frac-combine keeps values in [0,1) for full f32 precision.
// P[n][l] = frac( sum of f0_up/SR over frames 0..l-1 )   (exclusive)
// =====================================================================
__global__ void __launch_bounds__(1024)
phase_scan_kernel(const float* __restrict__ f0, float* __restrict__ P) {
  __shared__ float s[1024];
  const int n = blockIdx.x;
  const int l = threadIdx.x;

  float F = 0.0f;
  if (l < L_FRAMES) {
    const float* f = f0 + n * L_FRAMES;
    float B = f[l];
    float A = (l > 0)            ? f[l - 1] : B;
    float C = (l < L_FRAMES - 1) ? f[l + 1] : B;
    F = (30.0f * A + 180.0f * B + 30.0f * C) * INV_SR;  // ~[0.8, 3.8]
  }
  s[l] = F;
  __syncthreads();

  #pragma unroll
  for (int off = 1; off < 1024; off <<= 1) {
    float v = s[l];
    float u = (l >= off) ? s[l - off] : 0.0f;
    __syncthreads();
    float x = v + u;
    s[l] = x - truncf(x);          // frac-combine (all values >= 0)
    __syncthreads();
  }

  if (l < L_FRAMES) {
    P[n * L_FRAMES + l] = (l == 0) ? 0.0f : s[l - 1];
  }
}

// =====================================================================
// Kernel 2: fused upsample + phase + sin + noise mix.
//   out[n,t] = mask*Sum_h w~_h*sin(2pi*frac((h+1)*S_t + phi_h))
//            + (0.06*mask + 0.333*(1-mask)) * Sum_h w~_h * noise[n,h,t]
// S_t = P[n][l] + closed-form in-frame inclusive sum / SR.
// Noise staged via async global->LDS B128 while phases are computed.
// =====================================================================
__global__ void __launch_bounds__(TPB)
harmonic_kernel(const float* __restrict__ f0,
                const float* __restrict__ uv,
                const float* __restrict__ weights,
                const float* __restrict__ phi,
                const float* __restrict__ noise,
                const float* __restrict__ P,
                float* __restrict__ out) {
  __shared__ float4 tile[NHARM][TPB];   // 24 KB

  const int tid = threadIdx.x;
  const int n   = blockIdx.y;
  const int t0  = (blockIdx.x * TPB + tid) * SPT;

  // ---- issue async noise staging first (overlaps with ALU below) ----
  const float*  nbase = noise + ((long long)n * NHARM) * (long long)T_TOTAL + t0;
  const uint32_t lds0 = (uint32_t)(uintptr_t)(&tile[0][tid]);
  #pragma unroll
  for (int h = 0; h < NHARM; ++h) {
    async_ld_b128(lds0 + (uint32_t)(h * (TPB * 16)),
                  nbase + (long long)h * T_TOTAL);
  }

  // ---- normalized weights + phases (uniform -> scalar loads) ----
  float w[NHARM], ph[NHARM], s2 = 0.0f;
  #pragma unroll
  for (int h = 0; h < NHARM; ++h) {
    w[h]  = weights[h];
    ph[h] = phi[h];
    s2   += w[h] * w[h];
  }
  const float inv_norm = 1.0f / fmaxf(sqrtf(s2), 1e-12f);

  // ---- per-frame scalars (all 4 samples share one frame) ----
  const int l  = t0 / FRAME;
  const int j0 = t0 % FRAME;
  const float* f = f0 + n * L_FRAMES;
  const float B = f[l];
  const float A = (l > 0)            ? f[l - 1] : B;
  const float C = (l < L_FRAMES - 1) ? f[l + 1] : B;
  const float mask = uv[n * L_FRAMES + l];
  const float Pf   = P[n * L_FRAMES + l];

  // ---- closed-form inclusive in-frame sums -> phase base per sample ----
  float base[SPT];
  if (j0 < FRAME / 2) {              // ramp A->B region (j in [0,120))
    #pragma unroll
    for (int k = 0; k < SPT; ++k) {
      float jp1  = (float)(j0 + k + 1);
      float incl = jp1 * A + (B - A) * jp1 * (float)(j0 + k + 241) * (1.0f / 480.0f);
      base[k] = Pf + incl * INV_SR;
    }
  } else {                           // ramp B->C region (j in [120,240))
    #pragma unroll
    for (int k = 0; k < SPT; ++k) {
      float jp   = (float)(j0 + k - 119);    // j' + 1
      float incl = 30.0f * A + 90.0f * B + jp * B
                 + (C - B) * jp * jp * (1.0f / 480.0f);
      base[k] = Pf + incl * INV_SR;
    }
  }

  // ---- weighted harmonic sum (v_sin_f32; overlaps the async DMA) ----
  float sinsum[SPT];
  #pragma unroll
  for (int k = 0; k < SPT; ++k) {
    float acc = 0.0f;
    #pragma unroll
    for (int h = 0; h < NHARM; ++h) {
      float x = (float)(h + 1) * base[k] + ph[h];
      x = x - truncf(x);                       // frac, x >= 0
      acc += w[h] * __sinf(TWO_PI_F * x);
    }
    sinsum[k] = acc;
  }

  // ---- drain DMA, reduce noise channels from LDS ----
  wait_async0();
  float4 na = make_float4(0.0f, 0.0f, 0.0f, 0.0f);
  #pragma unroll
  for (int h = 0; h < NHARM; ++h) {
    float4 v = tile[h][tid];                   // ds_load_b128
    na.x += w[h] * v.x;  na.y += w[h] * v.y;
    na.z += w[h] * v.z;  na.w += w[h] * v.w;
  }

  const float c = mask * 0.06f + (1.0f - mask) * 0.333f;
  float4 o;
  o.x = inv_norm * (mask * sinsum[0] + c * na.x);
  o.y = inv_norm * (mask * sinsum[1] + c * na.y);
  o.z = inv_norm * (mask * sinsum[2] + c * na.z);
  o.w = inv_norm * (mask * sinsum[3] + c * na.w);

  *(float4*)(out + (long long)n * T_TOTAL + t0) = o;   // global_store_b128
}

// =====================================================================
extern "C" void kernel_launch(void* const* d_in, const int* in_sizes, int n_in,
                              void* d_out, int out_size, void* d_ws, size_t ws_size,
                              hipStream_t stream) {
  const float* f0      = (const float*)d_in[0];   // [32,1,800]
  const float* uv      = (const float*)d_in[1];   // [32,1,800]
  const float* weights = (const float*)d_in[2];   // [1,8,1]
  const float* phi     = (const float*)d_in[3];   // [1,8,1]
  const float* noise   = (const float*)d_in[4];   // [32,8,192000]
  float* out = (float*)d_out;                     // [32,1,192000]
  float* P   = (float*)d_ws;                      // 32*800 floats = 100 KB

  phase_scan_kernel<<<dim3(NBATCH), dim3(1024), 0, stream>>>(f0, P);

  dim3 grid(T_TOTAL / TILE, NBATCH);              // (250, 32)
  harmonic_kernel<<<grid, dim3(TPB), 0, stream>>>(f0, uv, weights, phi, noise, P, out);
}